// GNNRoutingPolicy_67053029425807
// MI455X (gfx1250) — compile-verified
//
#include <hip/hip_runtime.h>

#define N_ 100000
#define E_ 1000000
#define ET 1100000            // edges + self loops
#define CH 64                 // channels per head
#define HC 128                // H*C
#define XPAD 72               // padded node-feature stride (64 feat + 5 ctx + 3 zero)
#define SLOPE_ 0.2f
#define EPS_ 1e-5f

typedef __attribute__((ext_vector_type(2))) float v2f;
typedef __attribute__((ext_vector_type(8))) float v8f;

__device__ __forceinline__ float fatomic_add(float* p, float v) {
  return __hip_atomic_fetch_add(p, v, __ATOMIC_RELAXED, __HIP_MEMORY_SCOPE_AGENT);
}
// float atomic max via signed-max / unsigned-min trick (native integer atomics, no CAS loop)
__device__ __forceinline__ void fatomic_max(float* p, float v) {
  if (v >= 0.f)
    __hip_atomic_fetch_max((int*)p, __float_as_int(v), __ATOMIC_RELAXED, __HIP_MEMORY_SCOPE_AGENT);
  else
    __hip_atomic_fetch_min((unsigned int*)p, __float_as_uint(v), __ATOMIC_RELAXED, __HIP_MEMORY_SCOPE_AGENT);
}

__global__ void fill_kernel(float* __restrict__ p, float v, long n) {
  long i = (long)blockIdx.x * blockDim.x + threadIdx.x;
  if (i < n) p[i] = v;
}

// ---------------- WMMA f32 GEMM ----------------
// C[M x (16*nct)] = A[M x K] @ B[K x (16*nct)]  (+ optional bias, relu)
// One wave per 16x16 tile, 4 waves per block. M = 100000 = 6250*16 exactly.
// A frag (16x4 f32): lane<16 holds row m, K=k..k+1 ; lane>=16 holds row m, K=k+2..k+3
// B frag (4x16 f32): mirrored; C/D: VGPR r -> row r (+8 for upper half-wave), col = lane&15
__global__ void wmma_gemm(const float* __restrict__ A, int lda,
                          const float* __restrict__ B, int ldb,
                          float* __restrict__ C, int ldc,
                          const float* __restrict__ bias,
                          int K, int nct, int relu) {
  int wave = threadIdx.x >> 5;
  int lane = threadIdx.x & 31;
  int tile = blockIdx.x * 4 + wave;
  int rt = tile / nct, ct = tile % nct;
  int m  = lane & 15;
  int kq = (lane >> 4) << 1;            // 0 or 2
  long row  = (long)rt * 16 + m;
  int  colb = ct * 16;
  const float* Ap = A + row * lda + kq;
  const float* Bp = B + (long)kq * ldb + colb + m;
  v8f acc = {0.f, 0.f, 0.f, 0.f, 0.f, 0.f, 0.f, 0.f};
  for (int k = 0; k < K; k += 4) {
    v2f a, b;
    a.x = Ap[k];
    a.y = Ap[k + 1];
    b.x = Bp[(long)k * ldb];
    b.y = Bp[(long)(k + 1) * ldb];
    acc = __builtin_amdgcn_wmma_f32_16x16x4_f32(false, a, false, b, (short)0, acc, false, false);
  }
  long rbase = (long)rt * 16 + ((lane >> 4) << 3);
  int  cg = colb + m;
  float bv = bias ? bias[cg] : 0.f;
#pragma unroll
  for (int r = 0; r < 8; ++r) {
    float v = acc[r] + bv;
    if (relu) v = fmaxf(v, 0.f);
    C[(rbase + r) * ldc + cg] = v;
  }
}

// ---------------- attention scalars ----------------
__global__ void att_dots(const float* __restrict__ h, const float* __restrict__ as,
                         const float* __restrict__ ad, float* __restrict__ a_s,
                         float* __restrict__ a_d) {
  int n = blockIdx.x * blockDim.x + threadIdx.x;
  if (n >= N_) return;
  const float* hp = h + (long)n * HC;
  float s0 = 0.f, s1 = 0.f, d0 = 0.f, d1 = 0.f;
  for (int c = 0; c < CH; ++c) {
    float h0 = hp[c], h1 = hp[CH + c];
    s0 += h0 * as[c];      s1 += h1 * as[CH + c];
    d0 += h0 * ad[c];      d1 += h1 * ad[CH + c];
  }
  a_s[n * 2] = s0; a_s[n * 2 + 1] = s1;
  a_d[n * 2] = d0; a_d[n * 2 + 1] = d1;
}

__device__ __forceinline__ void edge_sd(const int* __restrict__ ei, int i, int& s, int& d) {
  if (i < E_) { s = ei[i]; d = ei[E_ + i]; }
  else        { s = i - E_; d = s; }          // self loop
}

// ---------------- edge pass 1: segment max ----------------
__global__ void edge_max(const int* __restrict__ ei, const float* __restrict__ a_s,
                         const float* __restrict__ a_d, float* __restrict__ emax) {
  int i = blockIdx.x * blockDim.x + threadIdx.x;
  if (i >= ET) return;
  int s, d; edge_sd(ei, i, s, d);
#pragma unroll
  for (int hh = 0; hh < 2; ++hh) {
    float v = a_s[s * 2 + hh] + a_d[d * 2 + hh];
    v = v > 0.f ? v : SLOPE_ * v;
    fatomic_max(&emax[d * 2 + hh], v);
  }
}

// ---------------- edge pass 2: exp + segment sum ----------------
__global__ void edge_expsum(const int* __restrict__ ei, const float* __restrict__ a_s,
                            const float* __restrict__ a_d, const float* __restrict__ emax,
                            float* __restrict__ ee, float* __restrict__ esum) {
  int t = blockIdx.x * blockDim.x + threadIdx.x;
  if (t >= ET * 2) return;
  int i = t >> 1, hh = t & 1;
  int s, d; edge_sd(ei, i, s, d);
  float v = a_s[s * 2 + hh] + a_d[d * 2 + hh];
  v = v > 0.f ? v : SLOPE_ * v;
  float x = __expf(v - emax[d * 2 + hh]);
  ee[t] = x;                                  // t == i*2+hh
  fatomic_add(&esum[d * 2 + hh], x);
}

// ---------------- edge pass 3: scatter messages ----------------
// thread per (edge, 4 channels): float4 gather of h[src], 8 native f32 atomics
__global__ void edge_scatter(const int* __restrict__ ei, const float* __restrict__ ee,
                             const float* __restrict__ h, float* __restrict__ acc) {
  long t = (long)blockIdx.x * blockDim.x + threadIdx.x;
  if (t >= (long)ET * 16) return;
  int i  = (int)(t >> 4);
  int c4 = ((int)t & 15) << 2;
  int s, d; edge_sd(ei, i, s, d);
  float e0 = ee[i * 2], e1 = ee[i * 2 + 1];
  const float4 h0 = *(const float4*)(h + (long)s * HC + c4);
  const float4 h1 = *(const float4*)(h + (long)s * HC + CH + c4);
  float* a0 = acc + (long)d * HC + c4;
  fatomic_add(a0 + 0, e0 * h0.x); fatomic_add(a0 + 1, e0 * h0.y);
  fatomic_add(a0 + 2, e0 * h0.z); fatomic_add(a0 + 3, e0 * h0.w);
  float* a1 = a0 + CH;
  fatomic_add(a1 + 0, e1 * h1.x); fatomic_add(a1 + 1, e1 * h1.y);
  fatomic_add(a1 + 2, e1 * h1.z); fatomic_add(a1 + 3, e1 * h1.w);
}

// ---------------- normalize + head-mean + bias, accumulate BN stats (LDS reduce) ----------------
__global__ void finalize_layer(const float* __restrict__ acc, const float* __restrict__ esum,
                               const float* __restrict__ bias, float* __restrict__ outp,
                               float* __restrict__ stats) {
  __shared__ float ls[CH], lq[CH];
  int tid = threadIdx.x;                       // 256 threads = 4 nodes x 64 channels
  if (tid < CH) { ls[tid] = 0.f; lq[tid] = 0.f; }
  __syncthreads();
  int c = tid & 63;
  int n = blockIdx.x * 4 + (tid >> 6);
  float v = 0.f;
  if (n < N_) {
    float s0 = esum[n * 2] + 1e-16f;
    float s1 = esum[n * 2 + 1] + 1e-16f;
    v = 0.5f * (acc[(long)n * HC + c] / s0 + acc[(long)n * HC + CH + c] / s1) + bias[c];
    outp[(long)n * XPAD + c] = v;
  }
  __hip_atomic_fetch_add(&ls[c], v,     __ATOMIC_RELAXED, __HIP_MEMORY_SCOPE_WORKGROUP);
  __hip_atomic_fetch_add(&lq[c], v * v, __ATOMIC_RELAXED, __HIP_MEMORY_SCOPE_WORKGROUP);
  __syncthreads();
  if (tid < CH) {
    fatomic_add(&stats[tid],      ls[tid]);
    fatomic_add(&stats[CH + tid], lq[tid]);
  }
}

__global__ void bn_relu(float* __restrict__ outp, const float* __restrict__ stats,
                        const float* __restrict__ g, const float* __restrict__ be) {
  long t = (long)blockIdx.x * blockDim.x + threadIdx.x;
  if (t >= (long)N_ * CH) return;
  int c  = (int)(t & 63);
  long n = t >> 6;
  float m   = stats[c]      * (1.f / N_);
  float var = stats[CH + c] * (1.f / N_) - m * m;
  float r = rsqrtf(var + EPS_);
  float v = (outp[n * XPAD + c] - m) * r * g[c] + be[c];
  outp[n * XPAD + c] = fmaxf(v, 0.f);
}

__global__ void write_ctx(const float* __restrict__ x, float* __restrict__ xa) {
  int n = blockIdx.x * blockDim.x + threadIdx.x;
  if (n >= N_) return;
#pragma unroll
  for (int j = 0; j < 5; ++j)
    xa[(long)n * XPAD + CH + j] = x[n * 16 + 9 + j];
}

__global__ void pad_w(const float* __restrict__ w, float* __restrict__ wp) {
  int t = blockIdx.x * blockDim.x + threadIdx.x;
  if (t < 69 * 64) wp[t] = w[t];               // rows 69..71 stay zero
}

__global__ void heads_kernel(const float* __restrict__ hm, const float* __restrict__ pW,
                             const float* __restrict__ pb, const float* __restrict__ vW,
                             const float* __restrict__ vb, float* __restrict__ out) {
  int n = blockIdx.x * blockDim.x + threadIdx.x;
  if (n >= N_) return;
  const float* hp = hm + (long)n * XPAD;
  float l = 0.f, v = 0.f;
  for (int c = 0; c < CH; ++c) { float hv = hp[c]; l += hv * pW[c]; v += hv * vW[c]; }
  out[n]      = l + pb[0];
  out[N_ + n] = v + vb[0];
}

extern "C" void kernel_launch(void* const* d_in, const int* in_sizes, int n_in,
                              void* d_out, int out_size, void* d_ws, size_t ws_size,
                              hipStream_t stream) {
  (void)in_sizes; (void)n_in; (void)out_size; (void)ws_size;
  const float* x  = (const float*)d_in[0];
  const int*   ei = (const int*)d_in[1];
  const float* W[3]  = {(const float*)d_in[2],  (const float*)d_in[8],  (const float*)d_in[14]};
  const float* As[3] = {(const float*)d_in[3],  (const float*)d_in[9],  (const float*)d_in[15]};
  const float* Ad[3] = {(const float*)d_in[4],  (const float*)d_in[10], (const float*)d_in[16]};
  const float* Bb[3] = {(const float*)d_in[5],  (const float*)d_in[11], (const float*)d_in[17]};
  const float* G[3]  = {(const float*)d_in[6],  (const float*)d_in[12], (const float*)d_in[18]};
  const float* Be[3] = {(const float*)d_in[7],  (const float*)d_in[13], (const float*)d_in[19]};
  const float* mW1 = (const float*)d_in[20];
  const float* mb1 = (const float*)d_in[21];
  const float* mW2 = (const float*)d_in[22];
  const float* mb2 = (const float*)d_in[23];
  const float* pW  = (const float*)d_in[24];
  const float* pb  = (const float*)d_in[25];
  const float* vW  = (const float*)d_in[26];
  const float* vb  = (const float*)d_in[27];

  float* ws   = (float*)d_ws;
  float* h    = ws;                              // N*128
  float* a_s  = h    + (size_t)N_ * HC;          // N*2
  float* a_d  = a_s  + (size_t)N_ * 2;           // N*2
  float* emax = a_d  + (size_t)N_ * 2;           // N*2
  float* esum = emax + (size_t)N_ * 2;           // N*2
  float* ee   = esum + (size_t)N_ * 2;           // ET*2
  float* acc  = ee   + (size_t)ET * 2;           // N*128
  float* xa   = acc  + (size_t)N_ * HC;          // N*72
  float* xb   = xa   + (size_t)N_ * XPAD;        // N*72
  float* wpad = xb   + (size_t)N_ * XPAD;        // 72*64
  float* stats = wpad + 72 * 64;                 // 128

  auto fill = [&](float* p, float v, long n) {
    fill_kernel<<<(int)((n + 255) / 256), 256, 0, stream>>>(p, v, n);
  };

  // one-time zeroing of padded ping-pong buffers + padded MLP weight
  fill(xa, 0.f, (long)N_ * XPAD);
  fill(xb, 0.f, (long)N_ * XPAD);
  fill(wpad, 0.f, 72 * 64);
  pad_w<<<(69 * 64 + 255) / 256, 256, 0, stream>>>(mW1, wpad);

  const int ROWT = N_ / 16;                      // 6250 row tiles (exact)
  auto gemm = [&](const float* A, int lda, const float* B, int ldb, float* Cp, int ldc,
                  const float* bias, int K, int ncols, int relu) {
    int nct = ncols / 16;
    int tiles = ROWT * nct;                      // divisible by 4 for ncols 64/128
    wmma_gemm<<<tiles / 4, 128, 0, stream>>>(A, lda, B, ldb, Cp, ldc, bias, K, nct, relu);
  };

  const float* cur = x; int curlda = 16, curK = 16;
  float* outbuf[3] = {xa, xb, xa};
  for (int L = 0; L < 3; ++L) {
    gemm(cur, curlda, W[L], HC, h, HC, nullptr, curK, HC, 0);
    att_dots<<<(N_ + 255) / 256, 256, 0, stream>>>(h, As[L], Ad[L], a_s, a_d);
    fill(emax, -3.402823466e38f, (long)N_ * 2);
    fill(esum, 0.f, (long)N_ * 2);
    fill(acc, 0.f, (long)N_ * HC);
    fill(stats, 0.f, 128);
    edge_max<<<(ET + 255) / 256, 256, 0, stream>>>(ei, a_s, a_d, emax);
    edge_expsum<<<(ET * 2 + 255) / 256, 256, 0, stream>>>(ei, a_s, a_d, emax, ee, esum);
    edge_scatter<<<(int)(((long)ET * 16 + 255) / 256), 256, 0, stream>>>(ei, ee, h, acc);
    finalize_layer<<<(N_ + 3) / 4, 256, 0, stream>>>(acc, esum, Bb[L], outbuf[L], stats);
    bn_relu<<<(int)(((long)N_ * CH + 255) / 256), 256, 0, stream>>>(outbuf[L], stats, G[L], Be[L]);
    cur = outbuf[L]; curlda = XPAD; curK = CH;
  }

  // MLP head: concat ctx into xa (cols 64..68; 69..71 already zero)
  write_ctx<<<(N_ + 255) / 256, 256, 0, stream>>>(x, xa);
  gemm(xa, XPAD, wpad, CH, xb, XPAD, mb1, XPAD, CH, 1);   // relu(h @ mW1 + mb1)
  gemm(xb, XPAD, mW2, CH, xa, XPAD, mb2, CH, CH, 0);      // h @ mW2 + mb2
  heads_kernel<<<(N_ + 255) / 256, 256, 0, stream>>>(xa, pW, pb, vW, vb, (float*)d_out);
}